// Custom_Softmax_48395691491506
// MI455X (gfx1250) — compile-verified
//
#include <hip/hip_runtime.h>
#include <stdint.h>
#include <stddef.h>

#define AS1 __attribute__((address_space(1)))
#define AS3 __attribute__((address_space(3)))

typedef int   vint4   __attribute__((ext_vector_type(4)));
typedef float vfloat4 __attribute__((ext_vector_type(4)));

// ---------------- problem constants ----------------
constexpr int S_LEN  = 1024;              // seq len (row length)
constexpr int NUM_H  = 16;
constexpr int NUM_B  = 4;
constexpr int THREADS = 128;              // 4 wave32s per block
constexpr int ROWS_PER_CHUNK = 4;         // one row per wave
constexpr int CHUNK_BYTES = ROWS_PER_CHUNK * S_LEN * 4;     // 16 KB (int32 elements)
constexpr int PIECES = CHUNK_BYTES / (THREADS * 16);        // 8 async b128 issues / thread
constexpr int TOTAL_ROWS   = NUM_B * NUM_H * S_LEN;         // 65536
constexpr int TOTAL_CHUNKS = TOTAL_ROWS / ROWS_PER_CHUNK;   // 16384
constexpr int CHUNKS_PER_BLOCK = 4;
constexpr int GRID = TOTAL_CHUNKS / CHUNKS_PER_BLOCK;       // 4096 blocks
static_assert(PIECES == 8, "wait-count literals below assume 8 pieces per chunk");

// ---------------- CDNA5 async global->LDS path ----------------
#if defined(__gfx1250__)
#  if __has_builtin(__builtin_amdgcn_global_load_async_to_lds_b128)
#    define ASYNC_MODE 1
#  else
#    define ASYNC_MODE 2
#  endif
#else
#  define ASYNC_MODE 0
#endif

#if ASYNC_MODE
#  if __has_builtin(__builtin_amdgcn_s_wait_asynccnt)
#    define WAIT_ASYNC_8() __builtin_amdgcn_s_wait_asynccnt(8)
#    define WAIT_ASYNC_0() __builtin_amdgcn_s_wait_asynccnt(0)
#  else
#    define WAIT_ASYNC_8() asm volatile("s_wait_asynccnt 8" ::: "memory")
#    define WAIT_ASYNC_0() asm volatile("s_wait_asynccnt 0" ::: "memory")
#  endif
#else
#  define WAIT_ASYNC_8() ((void)0)
#  define WAIT_ASYNC_0() ((void)0)
#endif

__device__ __forceinline__ void async_copy_b128(const void* gsrc, void* ldst) {
#if ASYNC_MODE == 1
  // builtin signature: (v4i AS1* gsrc, v4i AS3* ldst, imm offset, imm cpol)
  __builtin_amdgcn_global_load_async_to_lds_b128(
      (AS1 vint4*)(uintptr_t)gsrc,
      (AS3 vint4*)(uint32_t)(uintptr_t)ldst,  // low 32 bits of generic LDS addr = LDS offset
      0, 0);
#elif ASYNC_MODE == 2
  asm volatile("global_load_async_to_lds_b128 %0, %1, off"
               :: "v"((uint32_t)(uintptr_t)ldst),
                  "v"((uint64_t)(uintptr_t)gsrc)
               : "memory");
#else
  *(vint4*)ldst = *(const vint4*)gsrc;        // non-gfx1250 fallback (host pass)
#endif
}

__device__ __forceinline__ void copy_chunk(const char* gbase, char* lbase, int tid) {
#pragma unroll
  for (int p = 0; p < PIECES; ++p) {
    const int off = p * (THREADS * 16) + tid * 16;
    async_copy_b128(gbase + off, lbase + off);
  }
}

// ---------------- fused kernel ----------------
__global__ __launch_bounds__(THREADS)
void fused_int8_causal_softmax_requant(const int* __restrict__ x_q,
                                       const float* __restrict__ scale_x,
                                       const float* __restrict__ scale_out,
                                       float* __restrict__ out) {
  __shared__ alignas(16) char smem[2 * CHUNK_BYTES];   // 32 KB double buffer

  const int tid  = threadIdx.x;
  const int wave = tid >> 5;
  const int lane = tid & 31;
  const int chunk0 = blockIdx.x * CHUNKS_PER_BLOCK;
  const char* gsrc = (const char*)x_q;

  // prologue: stage chunk 0
  copy_chunk(gsrc + (size_t)chunk0 * CHUNK_BYTES, smem, tid);

  for (int i = 0; i < CHUNKS_PER_BLOCK; ++i) {
    const int chunk = chunk0 + i;
    char* cur = smem + (i & 1) * CHUNK_BYTES;

    // issue next chunk's async copy, then wait for the current chunk
    if (i + 1 < CHUNKS_PER_BLOCK) {
      copy_chunk(gsrc + (size_t)(chunk + 1) * CHUNK_BYTES,
                 smem + ((i + 1) & 1) * CHUNK_BYTES, tid);
      WAIT_ASYNC_8();     // in-order: <=8 outstanding means chunk i fully landed
    } else {
      WAIT_ASYNC_0();
    }
    __syncthreads();      // cross-wave visibility of the staged chunk

    // ---- one row per wave ----
    const int row = chunk * ROWS_PER_CHUNK + wave;        // global row in [0, 65536)
    const int r   = row & (S_LEN - 1);                    // row within S (causal bound)
    const int h   = (row >> 10) & (NUM_H - 1);            // head index
    const float sx = scale_x[(h << 10) + r];
    const float so = scale_out[(h << 10) + r];
    const int* lrow = (const int*)(cur + wave * (S_LEN * 4));

    // pass 1: dequant + causal mask + running max (32 values/lane in registers)
    float v[32];
    float m = -__builtin_inff();
#pragma unroll
    for (int j = 0; j < 8; ++j) {
      const int cbase = j * 128 + lane * 4;               // unit-stride across lanes
      const vint4 q = *(const vint4*)(lrow + cbase);      // ds_load_b128
#pragma unroll
      for (int k = 0; k < 4; ++k) {
        const int c = cbase + k;
        const float x = (c <= r) ? (float)q[k] * sx : -__builtin_inff();
        v[j * 4 + k] = x;
        m = fmaxf(m, x);
      }
    }
#pragma unroll
    for (int off = 16; off > 0; off >>= 1)
      m = fmaxf(m, __shfl_xor(m, off, 32));               // wave32 max

    // pass 2: exp + sum (exp(-inf - m) == 0 handles the masked tail)
    float s = 0.0f;
#pragma unroll
    for (int jj = 0; jj < 32; ++jj) {
      const float e = __expf(v[jj] - m);
      v[jj] = e;
      s += e;
    }
#pragma unroll
    for (int off = 16; off > 0; off >>= 1)
      s += __shfl_xor(s, off, 32);                        // wave32 sum

    // pass 3: requant p/so = e / (s*so), round-half-even, clamp, streaming store
    const float inv = 1.0f / (s * so);
    float* orow = out + (size_t)row * S_LEN;
#pragma unroll
    for (int j = 0; j < 8; ++j) {
      const int cbase = j * 128 + lane * 4;
      vfloat4 o;
#pragma unroll
      for (int k = 0; k < 4; ++k) {
        float p = rintf(v[j * 4 + k] * inv);
        o[k] = fminf(fmaxf(p, -128.0f), 127.0f);
      }
      __builtin_nontemporal_store(o, (vfloat4*)(orow + cbase));
    }

    __syncthreads();    // all reads of this buffer done before it is re-filled
  }
}

// ---------------- harness entry ----------------
extern "C" void kernel_launch(void* const* d_in, const int* in_sizes, int n_in,
                              void* d_out, int out_size, void* d_ws, size_t ws_size,
                              hipStream_t stream) {
  (void)in_sizes; (void)n_in; (void)out_size; (void)d_ws; (void)ws_size;
  const int*   x_q       = (const int*)d_in[0];
  const float* scale_x   = (const float*)d_in[1];
  const float* scale_out = (const float*)d_in[2];
  float*       out       = (float*)d_out;

  fused_int8_causal_softmax_requant<<<dim3(GRID), dim3(THREADS), 0, stream>>>(
      x_q, scale_x, scale_out, out);
}